// ObjCondensationLoss_38628935860479
// MI455X (gfx1250) — compile-verified
//
#include <hip/hip_runtime.h>
#include <hip/hip_bf16.h>
#include <math.h>

typedef __attribute__((ext_vector_type(2))) float v2f;
typedef __attribute__((ext_vector_type(8))) float v8f;

#define KMAX 1024   // upper bound on #clusters we support (reference uses 256)

struct OcWs {
    unsigned long long packed[KMAX];  // {beta_bits:32, ~index:32} per cluster
    float        bg_sum;              // sum of beta over background hits
    unsigned int bg_cnt;              // number of background hits
    float        pad0;
    float        pad1;
    float        xa[KMAX][4];         // condensation-point coords, padded (x,y,z,0)
    float        xan[KMAX];           // ||x_a||^2
    float        qak[KMAX];           // q_ak
};

// ---------------- Kernel 1: zero the accumulators -------------------------
__global__ void oc_init(OcWs* ws, float* out) {
    int t = blockIdx.x * blockDim.x + threadIdx.x;
    if (t == 0) { ws->bg_sum = 0.0f; ws->bg_cnt = 0u; out[0] = 0.0f; }
    for (int k = t; k < KMAX; k += gridDim.x * blockDim.x)
        ws->packed[k] = 0ull;
}

// ---------------- Kernel 2: per-hit reductions ----------------------------
// beta is monotone with q = atanh(beta)^2 + q_min, so a single packed
// argmax(beta) per cluster gives beta_ak, q_ak and idx_ak simultaneously.
// Pack {beta_bits, ~i}: atomicMax picks larger beta, and on exact ties the
// larger ~i == smaller i — identical to jnp.argmax's first-occurrence rule.
__global__ void oc_phase1(const float* __restrict__ beta,
                          const int*   __restrict__ y,
                          OcWs* ws, int N) {
    int i = blockIdx.x * blockDim.x + threadIdx.x;
    if (i >= N) return;
    float b  = beta[i];
    int   yi = y[i];
    if (yi < 0) {
        atomicAdd(&ws->bg_sum, b);
        atomicAdd(&ws->bg_cnt, 1u);
    } else if (yi < KMAX) {
        unsigned long long p = ((unsigned long long)__float_as_uint(b) << 32)
                             | (unsigned long long)(~(unsigned)i);
        atomicMax(&ws->packed[yi], p);
    }
}

// ---------------- Kernel 3: per-cluster prep + L_beta ---------------------
__global__ void oc_phase2(const float* __restrict__ x, OcWs* ws, float* out,
                          const int* Kp, const float* Sbp, const float* qminp) {
    int k = blockIdx.x * blockDim.x + threadIdx.x;
    int K = *Kp;
    if (k == 0) {
        // background term of L_beta
        float Nb = (float)max(ws->bg_cnt, 1u);
        atomicAdd(out, (*Sbp) * ws->bg_sum / Nb);
    }
    if (k >= K || k >= KMAX) return;

    unsigned long long p = ws->packed[k];
    float beta_ak; unsigned idx; float qak;
    if (p == 0ull) {                      // empty cluster: argmax of zeros -> idx 0
        beta_ak = 0.0f; idx = 0u; qak = 0.0f;
    } else {
        beta_ak = __uint_as_float((unsigned)(p >> 32));
        idx     = ~(unsigned)(p & 0xFFFFFFFFull);
        float t = 0.5f * __logf((1.0f + beta_ak) / (1.0f - beta_ak)); // atanh
        qak     = t * t + (*qminp);
    }
    float x0 = x[idx * 3 + 0], x1 = x[idx * 3 + 1], x2 = x[idx * 3 + 2];
    ws->xa[k][0] = x0; ws->xa[k][1] = x1; ws->xa[k][2] = x2; ws->xa[k][3] = 0.0f;
    ws->xan[k]   = x0 * x0 + x1 * x1 + x2 * x2;
    ws->qak[k]   = qak;
    atomicAdd(out, (1.0f - beta_ak) / (float)K);
}

// ---------------- Kernel 4: main N x K potential via WMMA -----------------
// Each wave owns a 16-row strip and sweeps all K columns in 16-wide tiles.
// One v_wmma_f32_16x16x4_f32 per tile produces the full 16x16 distance
// block:  A[m] = (-2x0,-2x1,-2x2, 1),  B[n] = (xa0,xa1,xa2,||xa||^2),
// C[m][n] = ||x_m||^2  =>  D[m][n] = ||x_m - xa_n||^2.
__global__ void __launch_bounds__(256)
oc_phase3(const float* __restrict__ x, const float* __restrict__ beta,
          const int* __restrict__ y, const OcWs* __restrict__ ws,
          float* out, const int* Kp, const float* qminp, int N, float invN) {
    __shared__ float s_xa[KMAX * 4];
    __shared__ float s_xan[KMAX];
    __shared__ float s_qak[KMAX];

    const int K     = *Kp;          // uniform scalar load
    const float qmn = *qminp;
    const int tid   = threadIdx.x;

    for (int k = tid; k < KMAX; k += 256) {
        if (k < K) {
            s_xa[4 * k + 0] = ws->xa[k][0];
            s_xa[4 * k + 1] = ws->xa[k][1];
            s_xa[4 * k + 2] = ws->xa[k][2];
            s_xa[4 * k + 3] = 0.0f;
            s_xan[k] = ws->xan[k];
            s_qak[k] = ws->qak[k];
        } else {
            s_xa[4 * k + 0] = 0.0f; s_xa[4 * k + 1] = 0.0f;
            s_xa[4 * k + 2] = 0.0f; s_xa[4 * k + 3] = 0.0f;
            s_xan[k] = 0.0f; s_qak[k] = 0.0f;   // tail tiles contribute 0
        }
    }
    __syncthreads();

    const int lane = tid & 31;
    const int wave = tid >> 5;
    const int half = lane >> 4;      // 0: K=0,1 lanes, 1: K=2,3 lanes
    const int lcol = lane & 15;
    const int row_base = (blockIdx.x * 8 + wave) * 16;

    // ---- A fragment (16x4 f32): lanes 0-15 hold K=0,1; lanes 16-31 K=2,3
    const int am  = row_base + lcol;
    const int amc = (am < N) ? am : (N - 1);
    v2f afrag;
    afrag.x = (half == 0) ? (-2.0f * x[amc * 3 + 0]) : (-2.0f * x[amc * 3 + 2]);
    afrag.y = (half == 0) ? (-2.0f * x[amc * 3 + 1]) : 1.0f;

    // ---- per-lane row metadata for the 8 accumulator rows
    int   yr[8]; float qir[8]; v8f c;
    #pragma unroll
    for (int r = 0; r < 8; ++r) {
        int  m  = row_base + r + 8 * half;
        bool ok = (m < N);
        int  mc = ok ? m : (N - 1);
        float b0 = x[mc * 3 + 0], b1 = x[mc * 3 + 1], b2 = x[mc * 3 + 2];
        c[r]  = b0 * b0 + b1 * b1 + b2 * b2;      // ||x_m||^2 seeds C
        yr[r] = ok ? y[mc] : -2;
        float bb = beta[mc];
        float t  = 0.5f * __logf((1.0f + bb) / (1.0f - bb));
        qir[r]   = ok ? (t * t + qmn) : 0.0f;     // q_i (0 kills OOB rows)
    }

    float s[8];
    #pragma unroll
    for (int r = 0; r < 8; ++r) s[r] = 0.0f;

    const int ntiles = (K + 15) >> 4;
    for (int t = 0; t < ntiles; ++t) {
        const int n = t * 16 + lcol;
        // B fragment (4x16 f32): lanes 0-15 hold K=0,1; lanes 16-31 K=2,3
        v2f bfrag;
        bfrag.x = (half == 0) ? s_xa[4 * n + 0] : s_xa[4 * n + 2];
        bfrag.y = (half == 0) ? s_xa[4 * n + 1] : s_xan[n];

        v8f d = __builtin_amdgcn_wmma_f32_16x16x4_f32(
            /*neg_a=*/false, afrag, /*neg_b=*/false, bfrag,
            /*c_mod=*/(short)0, c, /*reuse_a=*/false, /*reuse_b=*/false);

        const float q = s_qak[n];                 // q_ak for this column
        #pragma unroll
        for (int r = 0; r < 8; ++r) {
            float dd  = d[r];                     // ||x_m - xa_n||^2
            float pot = (yr[r] == n) ? dd : fmaxf(1.0f - dd, 0.0f);
            s[r] = fmaf(pot, q, s[r]);
        }
    }

    // weight by q_i, wave32 butterfly reduce, single atomic per wave
    float tot = 0.0f;
    #pragma unroll
    for (int r = 0; r < 8; ++r) tot = fmaf(s[r], qir[r], tot);
    #pragma unroll
    for (int off = 16; off > 0; off >>= 1)
        tot += __shfl_xor(tot, off, 32);
    if (lane == 0)
        atomicAdd(out, tot * invN);
}

// ---------------- host launcher -------------------------------------------
extern "C" void kernel_launch(void* const* d_in, const int* in_sizes, int n_in,
                              void* d_out, int out_size, void* d_ws, size_t ws_size,
                              hipStream_t stream) {
    const float* x     = (const float*)d_in[0];
    const float* beta  = (const float*)d_in[1];
    const int*   y     = (const int*)d_in[2];
    const int*   Kp    = (const int*)d_in[3];
    const float* Sbp   = (const float*)d_in[4];
    const float* qminp = (const float*)d_in[5];
    const int N = in_sizes[1];                // beta has N elements
    float* out = (float*)d_out;
    OcWs*  ws  = (OcWs*)d_ws;

    oc_init  <<<4, 256, 0, stream>>>(ws, out);
    oc_phase1<<<(N + 255) / 256, 256, 0, stream>>>(beta, y, ws, N);
    oc_phase2<<<KMAX / 256, 256, 0, stream>>>(x, ws, out, Kp, Sbp, qminp);

    const int strips = (N + 15) / 16;         // 16-row strips, 1 per wave
    const int blocks = (strips + 7) / 8;      // 8 waves (256 threads) per block
    oc_phase3<<<blocks, 256, 0, stream>>>(x, beta, y, ws, out, Kp, qminp, N,
                                          1.0f / (float)N);
}